// HGNN_30983894073780
// MI455X (gfx1250) — compile-verified
//
#include <hip/hip_runtime.h>
#include <hip/hip_bf16.h>

// ---------------------------------------------------------------------------
// Problem constants (match reference)
// ---------------------------------------------------------------------------
#define NV_USER 50000
#define NV_ITEM 50000
#define NEDGE   800000
#define DIN     256
#define DHID    256
#define DOUT    128

typedef float v2f __attribute__((ext_vector_type(2)));
typedef float v8f __attribute__((ext_vector_type(8)));

// ---------------------------------------------------------------------------
// zero fill (grid-stride)
// ---------------------------------------------------------------------------
__global__ void zero_f32(float* __restrict__ p, long n) {
    long i = (long)blockIdx.x * blockDim.x + threadIdx.x;
    long stride = (long)gridDim.x * blockDim.x;
    for (; i < n; i += stride) p[i] = 0.0f;
}

// ---------------------------------------------------------------------------
// degree accumulation: outdeg[src[e]] += 1 ; indeg[dst[e]] += 1
// ---------------------------------------------------------------------------
__global__ void degree_accum(const int* __restrict__ src,
                             const int* __restrict__ dst,
                             float* __restrict__ outdeg,
                             float* __restrict__ indeg, int E) {
    int i = blockIdx.x * blockDim.x + threadIdx.x;
    if (i < E) {
        unsafeAtomicAdd(&outdeg[src[i]], 1.0f);
        unsafeAtomicAdd(&indeg[dst[i]], 1.0f);
    }
}

// deg -> rsqrt(max(deg,1))  in place
__global__ void rnorm_inplace(float* __restrict__ p, long n) {
    long i = (long)blockIdx.x * blockDim.x + threadIdx.x;
    if (i < n) p[i] = rsqrtf(fmaxf(p[i], 1.0f));
}

// ---------------------------------------------------------------------------
// feat = (X * rn[row]) @ W   via V_WMMA_F32_16X16X4_F32
// one wave (32 threads) computes a 16x64 output tile; K stepped by 4.
// grid = (M/16, Nout/64), block = 32.  M % 16 == 0, Nout % 64 == 0.
// ---------------------------------------------------------------------------
__global__ __launch_bounds__(32) void gemm_rn_wmma(
    const float* __restrict__ X, const float* __restrict__ rn,
    const float* __restrict__ W, float* __restrict__ out,
    int M, int K, int Nout)
{
    const int lane = threadIdx.x & 31;
    const int lh   = lane >> 4;     // lane half: 0 or 1
    const int lm   = lane & 15;
    const int m0   = blockIdx.x * 16;
    const int n0   = blockIdx.y * 64;

    const int   row  = m0 + lm;                 // A rows: both halves hold M=0..15
    const float r    = rn[row];
    const float* xrow = X + (size_t)row * K;

    v8f acc0 = {}, acc1 = {}, acc2 = {}, acc3 = {};

    for (int k0 = 0; k0 < K; k0 += 4) {
        const int kb = k0 + 2 * lh;             // half-lane K offset (A & B)
        v2f a;
        a.x = xrow[kb]     * r;
        a.y = xrow[kb + 1] * r;

        const float* wr0 = W + (size_t)kb * Nout + n0 + lm;
        const float* wr1 = wr0 + Nout;
        v2f b0; b0.x = wr0[0];  b0.y = wr1[0];
        v2f b1; b1.x = wr0[16]; b1.y = wr1[16];
        v2f b2; b2.x = wr0[32]; b2.y = wr1[32];
        v2f b3; b3.x = wr0[48]; b3.y = wr1[48];

        acc0 = __builtin_amdgcn_wmma_f32_16x16x4_f32(false, a, false, b0, (short)0, acc0, false, false);
        acc1 = __builtin_amdgcn_wmma_f32_16x16x4_f32(false, a, false, b1, (short)0, acc1, false, false);
        acc2 = __builtin_amdgcn_wmma_f32_16x16x4_f32(false, a, false, b2, (short)0, acc2, false, false);
        acc3 = __builtin_amdgcn_wmma_f32_16x16x4_f32(false, a, false, b3, (short)0, acc3, false, false);
    }

    // C/D layout: VGPR v -> row m0 + v + 8*lh, col n0 + lm (+0/16/32/48)
    #pragma unroll
    for (int v = 0; v < 8; ++v) {
        const int m = m0 + v + 8 * lh;
        float* o = out + (size_t)m * Nout + n0 + lm;
        o[0]  = acc0[v];
        o[16] = acc1[v];
        o[32] = acc2[v];
        o[48] = acc3[v];
    }
}

// ---------------------------------------------------------------------------
// scatter: agg[dst[e], :] += feat[src[e], :] * rn_in[dst[e]] * scale
// (D/4) float4 chunks per edge; chunkShift = log2(D/4); blockDim = 256.
// ---------------------------------------------------------------------------
__global__ __launch_bounds__(256) void scatter_edges(
    const float* __restrict__ feat,
    const int* __restrict__ src, const int* __restrict__ dst,
    const float* __restrict__ rn_in, float scale,
    float* __restrict__ agg, int E, int D, int chunkShift)
{
    const int chunks = 1 << chunkShift;
    const int edges_per_block = blockDim.x >> chunkShift;
    const int e = blockIdx.x * edges_per_block + (threadIdx.x >> chunkShift);
    if (e >= E) return;
    const int c = threadIdx.x & (chunks - 1);

    const int s = src[e];
    const int d = dst[e];
    const float w = rn_in[d] * scale;

    const float4 f = *(const float4*)(feat + (size_t)s * D + c * 4);
    float* a = agg + (size_t)d * D + c * 4;
    unsafeAtomicAdd(a + 0, f.x * w);
    unsafeAtomicAdd(a + 1, f.y * w);
    unsafeAtomicAdd(a + 2, f.z * w);
    unsafeAtomicAdd(a + 3, f.w * w);
}

// ---------------------------------------------------------------------------
// out = act(agg + coef*(b1[j] + b2[j])),  j = i & (D-1), D power of 2
// ---------------------------------------------------------------------------
__global__ void bias_act(const float* __restrict__ agg,
                         const float* __restrict__ b1,
                         const float* __restrict__ b2,
                         float coef, float* __restrict__ out,
                         long n, int dmask, int do_relu)
{
    long i = (long)blockIdx.x * blockDim.x + threadIdx.x;
    if (i >= n) return;
    const int j = (int)(i & dmask);
    float v = agg[i] + coef * (b1[j] + b2[j]);
    out[i] = do_relu ? fmaxf(v, 0.0f) : v;
}

// ---------------------------------------------------------------------------
// Host-side orchestration
// ---------------------------------------------------------------------------
extern "C" void kernel_launch(void* const* d_in, const int* in_sizes, int n_in,
                              void* d_out, int out_size, void* d_ws, size_t ws_size,
                              hipStream_t stream) {
    const float* x_user = (const float*)d_in[0];
    const float* x_item = (const float*)d_in[1];
    const float* W1_f   = (const float*)d_in[2];
    const float* b1_f   = (const float*)d_in[3];
    const float* W1_r   = (const float*)d_in[4];
    const float* b1_r   = (const float*)d_in[5];
    const float* W1_rb  = (const float*)d_in[6];
    const float* b1_rb  = (const float*)d_in[7];
    const float* W2_f   = (const float*)d_in[8];
    const float* b2_f   = (const float*)d_in[9];
    const float* W2_r   = (const float*)d_in[10];
    const float* b2_r   = (const float*)d_in[11];
    const float* W2_rb  = (const float*)d_in[12];
    const float* b2_rb  = (const float*)d_in[13];
    const int* src_f    = (const int*)d_in[14];
    const int* dst_f    = (const int*)d_in[15];
    const int* src_r    = (const int*)d_in[16];
    const int* dst_r    = (const int*)d_in[17];
    const int* src_rb   = (const int*)d_in[18];
    const int* dst_rb   = (const int*)d_in[19];

    const long N  = NV_USER;          // == NV_ITEM
    const int  E  = NEDGE;

    // workspace layout (floats):
    //   deg/rnorm 6*N | feat N*256 | agg_u N*256 | agg_i N*256 | h_u N*256 | h_i N*256
    float* ws     = (float*)d_ws;
    float* deg    = ws;
    float* feat   = deg + 6 * N;
    float* agg_u  = feat  + N * (long)DHID;
    float* agg_i  = agg_u + N * (long)DHID;
    float* h_u    = agg_i + N * (long)DHID;
    float* h_i    = h_u   + N * (long)DHID;

    float* rs_out_f  = deg + 0 * N;   // rsqrt(out_deg) follows (users)
    float* rs_in_f   = deg + 1 * N;   // rsqrt(in_deg)  follows (users)
    float* rs_out_r  = deg + 2 * N;   // rates: users
    float* rs_in_r   = deg + 3 * N;   // rates: items
    float* rs_out_rb = deg + 4 * N;   // ratedby: items
    float* rs_in_rb  = deg + 5 * N;   // ratedby: users

    float* o_u = (float*)d_out;
    float* o_i = o_u + N * (long)DOUT;

    const int ZB = 2048;               // grid-stride zero blocks
    const int EB = (E + 255) / 256;    // per-edge blocks

    // --- degrees (edge-set dependent only) ---
    hipLaunchKernelGGL(zero_f32, dim3(256), dim3(256), 0, stream, deg, 6 * N);
    hipLaunchKernelGGL(degree_accum, dim3(EB), dim3(256), 0, stream, src_f,  dst_f,  rs_out_f,  rs_in_f,  E);
    hipLaunchKernelGGL(degree_accum, dim3(EB), dim3(256), 0, stream, src_r,  dst_r,  rs_out_r,  rs_in_r,  E);
    hipLaunchKernelGGL(degree_accum, dim3(EB), dim3(256), 0, stream, src_rb, dst_rb, rs_out_rb, rs_in_rb, E);
    hipLaunchKernelGGL(rnorm_inplace, dim3((int)((6 * N + 255) / 256)), dim3(256), 0, stream, deg, 6 * N);

    // ===================== layer 1 (D = 256) =====================
    hipLaunchKernelGGL(zero_f32, dim3(ZB), dim3(256), 0, stream, agg_u, N * (long)DHID);
    hipLaunchKernelGGL(zero_f32, dim3(ZB), dim3(256), 0, stream, agg_i, N * (long)DHID);

    const dim3 g1((int)(N / 16), DHID / 64), b32(32);
    const int sc1 = 6;                               // 256/4 = 64 chunks
    const int SB1 = E * (DHID / 4) / 256;            // scatter blocks

    // follows: user -> user (weight 0.5 into agg_u)
    hipLaunchKernelGGL(gemm_rn_wmma, g1, b32, 0, stream, x_user, rs_out_f, W1_f, feat, (int)N, DIN, DHID);
    hipLaunchKernelGGL(scatter_edges, dim3(SB1), dim3(256), 0, stream, feat, src_f, dst_f, rs_in_f, 0.5f, agg_u, E, DHID, sc1);
    // ratedby: item -> user (weight 0.5 into agg_u)
    hipLaunchKernelGGL(gemm_rn_wmma, g1, b32, 0, stream, x_item, rs_out_rb, W1_rb, feat, (int)N, DIN, DHID);
    hipLaunchKernelGGL(scatter_edges, dim3(SB1), dim3(256), 0, stream, feat, src_rb, dst_rb, rs_in_rb, 0.5f, agg_u, E, DHID, sc1);
    // rates: user -> item (weight 1.0 into agg_i)
    hipLaunchKernelGGL(gemm_rn_wmma, g1, b32, 0, stream, x_user, rs_out_r, W1_r, feat, (int)N, DIN, DHID);
    hipLaunchKernelGGL(scatter_edges, dim3(SB1), dim3(256), 0, stream, feat, src_r, dst_r, rs_in_r, 1.0f, agg_i, E, DHID, sc1);

    // h_u = relu(agg_u + 0.5*(b1_f + b1_rb)),  h_i = relu(agg_i + b1_r)
    const long n1 = N * (long)DHID;
    hipLaunchKernelGGL(bias_act, dim3((int)((n1 + 255) / 256)), dim3(256), 0, stream,
                       agg_u, b1_f, b1_rb, 0.5f, h_u, n1, DHID - 1, 1);
    hipLaunchKernelGGL(bias_act, dim3((int)((n1 + 255) / 256)), dim3(256), 0, stream,
                       agg_i, b1_r, b1_r, 0.5f, h_i, n1, DHID - 1, 1);

    // ===================== layer 2 (D = 128) =====================
    hipLaunchKernelGGL(zero_f32, dim3(ZB), dim3(256), 0, stream, agg_u, N * (long)DOUT);
    hipLaunchKernelGGL(zero_f32, dim3(ZB), dim3(256), 0, stream, agg_i, N * (long)DOUT);

    const dim3 g2((int)(N / 16), DOUT / 64);
    const int sc2 = 5;                               // 128/4 = 32 chunks
    const int SB2 = E * (DOUT / 4) / 256;

    hipLaunchKernelGGL(gemm_rn_wmma, g2, b32, 0, stream, h_u, rs_out_f, W2_f, feat, (int)N, DHID, DOUT);
    hipLaunchKernelGGL(scatter_edges, dim3(SB2), dim3(256), 0, stream, feat, src_f, dst_f, rs_in_f, 0.5f, agg_u, E, DOUT, sc2);
    hipLaunchKernelGGL(gemm_rn_wmma, g2, b32, 0, stream, h_i, rs_out_rb, W2_rb, feat, (int)N, DHID, DOUT);
    hipLaunchKernelGGL(scatter_edges, dim3(SB2), dim3(256), 0, stream, feat, src_rb, dst_rb, rs_in_rb, 0.5f, agg_u, E, DOUT, sc2);
    hipLaunchKernelGGL(gemm_rn_wmma, g2, b32, 0, stream, h_u, rs_out_r, W2_r, feat, (int)N, DHID, DOUT);
    hipLaunchKernelGGL(scatter_edges, dim3(SB2), dim3(256), 0, stream, feat, src_r, dst_r, rs_in_r, 1.0f, agg_i, E, DOUT, sc2);

    // o_u = agg_u + 0.5*(b2_f + b2_rb),  o_i = agg_i + b2_r  (no relu)
    const long n2 = N * (long)DOUT;
    hipLaunchKernelGGL(bias_act, dim3((int)((n2 + 255) / 256)), dim3(256), 0, stream,
                       agg_u, b2_f, b2_rb, 0.5f, o_u, n2, DOUT - 1, 0);
    hipLaunchKernelGGL(bias_act, dim3((int)((n2 + 255) / 256)), dim3(256), 0, stream,
                       agg_i, b2_r, b2_r, 0.5f, o_i, n2, DOUT - 1, 0);
}